// AdaptivePConv_39728447488763
// MI455X (gfx1250) — compile-verified
//
#include <hip/hip_runtime.h>

// ---------------- problem constants (from reference setup_inputs) ----------
constexpr int Bn   = 8;
constexpr int Cn   = 64;
constexpr int Hn   = 256;
constexpr int Wn   = 256;
constexpr int OCn  = 64;
constexpr int Pn   = 16;            // selected channels
constexpr int HWn  = Hn * Wn;       // 65536
constexpr int OUTC = OCn + Cn - Pn; // 112
constexpr int KKn  = Pn * 9;        // GEMM K = 144
constexpr int NQ   = KKn / 4;       // 36 K-quads
constexpr float EPS = 1e-8f;

typedef float v2f __attribute__((ext_vector_type(2)));
typedef float v8f __attribute__((ext_vector_type(8)));

// constexpr tap tables: k -> (channel p, tap index r=ky*3+kx, spatial offset)
struct TapTables {
    int p[KKn];   // k / 9
    int rc[KKn];  // (ky-1)*W + (kx-1)
    int idx[KKn]; // ky*3+kx  (bit position in validity mask)
    constexpr TapTables() : p(), rc(), idx() {
        for (int k = 0; k < KKn; ++k) {
            int pp = k / 9, r = k % 9, ky = r / 3, kx = r % 3;
            p[k] = pp;
            rc[k] = (ky - 1) * Wn + (kx - 1);
            idx[k] = r;
        }
    }
};
constexpr TapTables TT{};

__device__ __forceinline__ int csel(bool hi, int a, int b) { return hi ? a : b; }

// ---------------- kernel 1: per-channel min/max + histogram + entropy -----
__global__ __launch_bounds__(256)
void stats_kernel(const float* __restrict__ x, float* __restrict__ activity) {
    const int bc  = blockIdx.x;                 // b*C + c
    const int tid = threadIdx.x;
    const float4* __restrict__ base4 =
        reinterpret_cast<const float4*>(x + (size_t)bc * HWn);

    __shared__ float    s_min[256];
    __shared__ float    s_max[256];
    __shared__ unsigned s_hist[256];
    __shared__ float    s_red[256];

    // pass 1: min/max (float4 sweeps)
    float vmin =  3.402823466e38f;
    float vmax = -3.402823466e38f;
    for (int i = tid; i < HWn / 4; i += 256) {
        float4 v = base4[i];
        vmin = fminf(vmin, fminf(fminf(v.x, v.y), fminf(v.z, v.w)));
        vmax = fmaxf(vmax, fmaxf(fmaxf(v.x, v.y), fmaxf(v.z, v.w)));
    }
    s_min[tid] = vmin; s_max[tid] = vmax;
    __syncthreads();
    for (int s = 128; s > 0; s >>= 1) {
        if (tid < s) {
            s_min[tid] = fminf(s_min[tid], s_min[tid + s]);
            s_max[tid] = fmaxf(s_max[tid], s_max[tid + s]);
        }
        __syncthreads();
    }
    const float gmin  = s_min[0];
    const float scale = 256.0f / (s_max[0] - gmin + EPS);

    // pass 2: histogram (LDS integer atomics; channel stays hot in L2)
    s_hist[tid] = 0u;
    __syncthreads();
    for (int i = tid; i < HWn / 4; i += 256) {
        float4 v = base4[i];
        float vv[4] = {v.x, v.y, v.z, v.w};
        #pragma unroll
        for (int j = 0; j < 4; ++j) {
            int bin = (int)((vv[j] - gmin) * scale);
            bin = bin < 0 ? 0 : (bin > 255 ? 255 : bin);
            atomicAdd(&s_hist[bin], 1u);
        }
    }
    __syncthreads();

    // entropy: hist+eps, prob = h/sum, H = -sum p*log(p+eps)
    const float total = (float)HWn + 256.0f * EPS;
    float h = (float)s_hist[tid] + EPS;
    float p = h / total;
    s_red[tid] = -p * logf(p + EPS);
    __syncthreads();
    for (int s = 128; s > 0; s >>= 1) {
        if (tid < s) s_red[tid] += s_red[tid + s];
        __syncthreads();
    }
    if (tid == 0) activity[bc] = s_red[0];
}

// ---------------- kernel 2: top-16 + unselected (ascending) ---------------
__global__ void topk_kernel(const float* __restrict__ activity,
                            int* __restrict__ sel, int* __restrict__ unsel) {
    const int b = blockIdx.x;
    __shared__ float act[Cn];
    __shared__ int   taken[Cn];
    const int t = threadIdx.x;
    act[t]   = activity[b * Cn + t];
    taken[t] = 0;
    __syncthreads();
    if (t == 0) {
        for (int i = 0; i < Pn; ++i) {          // stable argmax (jax tie rule)
            float best = -3.402823466e38f; int bi = 0;
            for (int c = 0; c < Cn; ++c)
                if (!taken[c] && act[c] > best) { best = act[c]; bi = c; }
            taken[bi] = 1;
            sel[b * Pn + i] = bi;
        }
        int j = 0;
        for (int c = 0; c < Cn; ++c)
            if (!taken[c]) unsel[b * (Cn - Pn) + j++] = c;
    }
}

// ---------------- kernel 3: implicit-GEMM conv via fp32 WMMA --------------
// Each wave: 16 OC x 128 pixels (8 tiles of 16x16x(K=144) = 8 x 36 WMMAs).
// A-matrix (weights) preloaded to VGPRs once; im2col offsets precomputed;
// padding handled branchlessly via a 9-bit per-tile tap-validity mask.
__global__ __launch_bounds__(256)
void conv_wmma_kernel(const float* __restrict__ x,
                      const float* __restrict__ weight,
                      const float* __restrict__ bias,
                      const int*   __restrict__ sel,
                      float* __restrict__ out) {
    const int lane  = threadIdx.x & 31;
    const int wave  = threadIdx.x >> 5;           // 0..7
    const bool hi   = (lane >> 4) != 0;           // lanes 16-31 hold K+2,K+3
    const int kHalf = hi ? 2 : 0;
    const int b     = blockIdx.z;
    const int ocb   = blockIdx.y * 16;            // OC tile base
    const int waveBase = (blockIdx.x * 8 + wave) * 128;  // 128-px strip (1 row)

    const int mA    = lane & 15;                  // A-matrix row (OC)
    const int mBase = hi ? 8 : 0;                 // C/D row split

    const float* __restrict__ wrow = weight + (size_t)(ocb + mA) * KKn;
    const int*   __restrict__ selb = sel + b * Pn;
    const float* __restrict__ xb   = x + (size_t)b * Cn * HWn;

    // uniform per-block: selected-channel base offsets (scalarizes to SGPRs)
    int selOff[Pn];
    #pragma unroll
    for (int p = 0; p < Pn; ++p) selOff[p] = selb[p] * HWn;

    // preload A (16x144 weights) into registers + im2col offsets
    v2f aReg[NQ];
    int off0[NQ], off1[NQ];
    #pragma unroll
    for (int i = 0; i < NQ; ++i) {
        const int kk = i * 4;
        aReg[i].x = wrow[kHalf + kk];
        aReg[i].y = wrow[kHalf + kk + 1];
        off0[i] = csel(hi, selOff[TT.p[kk + 2]] + TT.rc[kk + 2],
                           selOff[TT.p[kk + 0]] + TT.rc[kk + 0]);
        off1[i] = csel(hi, selOff[TT.p[kk + 3]] + TT.rc[kk + 3],
                           selOff[TT.p[kk + 1]] + TT.rc[kk + 1]);
    }

    // bias for this wave's 8 accumulator rows
    float bVal[8];
    #pragma unroll
    for (int r = 0; r < 8; ++r) bVal[r] = bias[ocb + mBase + r];

    // h is uniform for the whole 128-pixel strip
    const int h  = waveBase >> 8;                 // / Wn
    const int mh = (h > 0 ? 1 : 0) | 8 | (h < Hn - 1 ? 64 : 0);  // uniform

    const size_t outBase = ((size_t)b * OUTC + ocb + mBase) * HWn;

    #pragma unroll 1
    for (int t = 0; t < 8; ++t) {
        const int n = waveBase + t * 16 + (lane & 15);   // output column/pixel
        const int w = n & (Wn - 1);
        const int row3 = (w > 0 ? 1 : 0) | 2 | (w < Wn - 1 ? 4 : 0);
        const int tm   = row3 * mh;                      // 9-bit tap validity
        const float* __restrict__ xn = xb + n;

        v8f acc = {};
        #pragma unroll
        for (int i = 0; i < NQ; ++i) {
            const int kk = i * 4;
            const int i0 = csel(hi, TT.idx[kk + 2], TT.idx[kk + 0]);
            const int i1 = csel(hi, TT.idx[kk + 3], TT.idx[kk + 1]);
            const bool v0 = (tm >> i0) & 1;
            const bool v1 = (tm >> i1) & 1;
            float b0 = xn[v0 ? off0[i] : -n];            // clamped, in-bounds
            float b1 = xn[v1 ? off1[i] : -n];
            v2f bm;
            bm.x = v0 ? b0 : 0.0f;
            bm.y = v1 ? b1 : 0.0f;
            acc = __builtin_amdgcn_wmma_f32_16x16x4_f32(
                false, aReg[i], false, bm, (short)0, acc, false, false);
        }

        #pragma unroll
        for (int r = 0; r < 8; ++r)
            out[outBase + (size_t)r * HWn + n] = acc[r] + bVal[r];
    }
}

// ---------------- kernel 4: copy unselected channels ----------------------
__global__ __launch_bounds__(256)
void copy_kernel(const float* __restrict__ x, const int* __restrict__ unsel,
                 float* __restrict__ out) {
    const int b = blockIdx.z;
    const int j = blockIdx.y;                          // 0..47
    const int base = (blockIdx.x * 256 + threadIdx.x) * 4;
    const int src_c = unsel[b * (Cn - Pn) + j];
    const float4* __restrict__ src = reinterpret_cast<const float4*>(
        x + ((size_t)(b * Cn + src_c)) * HWn + base);
    float4* __restrict__ dst = reinterpret_cast<float4*>(
        out + ((size_t)b * OUTC + OCn + j) * HWn + base);
    *dst = *src;
}

// ---------------- launcher -------------------------------------------------
extern "C" void kernel_launch(void* const* d_in, const int* in_sizes, int n_in,
                              void* d_out, int out_size, void* d_ws, size_t ws_size,
                              hipStream_t stream) {
    const float* x      = (const float*)d_in[0];
    const float* weight = (const float*)d_in[1];
    const float* bias   = (const float*)d_in[2];
    float* out = (float*)d_out;

    float* activity = (float*)d_ws;                      // B*C floats
    int*   sel      = (int*)((char*)d_ws + 4096);        // B*P ints
    int*   unsel    = (int*)((char*)d_ws + 8192);        // B*(C-P) ints

    stats_kernel<<<dim3(Bn * Cn), dim3(256), 0, stream>>>(x, activity);
    topk_kernel<<<dim3(Bn), dim3(Cn), 0, stream>>>(activity, sel, unsel);
    conv_wmma_kernel<<<dim3(HWn / 1024, OCn / 16, Bn), dim3(256), 0, stream>>>(
        x, weight, bias, sel, out);
    copy_kernel<<<dim3(HWn / 1024, Cn - Pn, Bn), dim3(256), 0, stream>>>(
        x, unsel, out);
}